// coarse2fine_module_50809463112196
// MI455X (gfx1250) — compile-verified
//
#include <hip/hip_runtime.h>
#include <cstdint>

typedef __attribute__((ext_vector_type(2))) float v2f;
typedef __attribute__((ext_vector_type(8))) float v8f;

#define ACT_NONE  0
#define ACT_LEAKY 1
#define ACT_RELU  2

// ---------------------------------------------------------------------------
// Input fetch for conv GEMM (im2col on the fly for KS=3, pad=1).
// k is a compile-time constant after full unroll -> ci/ky/kx constant-fold.
// ---------------------------------------------------------------------------
template<int KS>
__device__ __forceinline__ float load_in(const float* __restrict__ In, int k, int x, int h,
                                         int H, int W, int HW) {
  if (KS == 1) {
    return (x < W) ? In[k * HW + h * W + x] : 0.0f;
  } else {
    const int ci = k / 9;
    const int r  = k - ci * 9;
    const int ky = r / 3;
    const int kx = r - ky * 3;
    const int y  = h + ky - 1;
    const int xx = x + kx - 1;
    const bool ok = (y >= 0) & (y < H) & (xx >= 0) & (xx < W);
    return ok ? In[ci * HW + y * W + xx] : 0.0f;
  }
}

// ---------------------------------------------------------------------------
// Square conv (Cin == Cout == C) as GEMM with V_WMMA_F32_16X16X4_F32.
//   D[co][h][w] = sum_k Wt[co][k] * im2col(In)[k][h][w],  K = C*KS*KS
// One wave computes a 16(spatial w in one row h) x Cout tile; for C=24 it
// holds two accumulators (two 16-row weight tiles) and reuses each B fragment.
// Weights are staged in LDS once per block (zero-padded to COT*16 rows).
// Fragment layouts per ISA 7.12.2:
//   A 16x4 : lane = {M = lane&15}, VGPR0/1 hold K = 2*(lane>>4) + {0,1}
//   B 4x16 : VGPR v holds row K = v + 2*(lane>>4), col N = lane&15
//   C 16x16: VGPR r holds row M = r + 8*(lane>>4), col N = lane&15
// ---------------------------------------------------------------------------
template<int KS, int C>
__global__ void conv_wmma_kernel(const float* __restrict__ In,
                                 const float* __restrict__ Wt,
                                 float* __restrict__ Out,
                                 int H, int W) {
  constexpr int K   = C * KS * KS;
  constexpr int COT = (C + 15) / 16;       // 1 (C=12) or 2 (C=24)
  __shared__ float wlds[COT * 16 * K];

  // Cooperative weight stage (rows >= C zero-padded so A pad rows are 0)
  for (int i = threadIdx.x; i < COT * 16 * K; i += blockDim.x) {
    const int co = i / K;
    const int k  = i - co * K;
    wlds[i] = (co < C) ? Wt[co * K + k] : 0.0f;
  }
  __syncthreads();

  const int lane = threadIdx.x & 31;
  const int waveInBlock = threadIdx.x >> 5;
  const int wavesPerBlock = blockDim.x >> 5;
  const int tilesW = (W + 15) >> 4;
  const int HW = H * W;
  const int totalWaves = tilesW * H;
  const int gw = blockIdx.x * wavesPerBlock + waveInBlock;
  if (gw >= totalWaves) return;            // wave-uniform exit (after barrier)

  const int tw = gw % tilesW;
  const int h  = gw / tilesW;
  const int w0 = tw << 4;

  const int row = lane & 15;               // A: M / B,C: N index
  const int hi  = lane >> 4;               // half-wave select
  const int x   = w0 + row;

  v8f acc[COT];
  #pragma unroll
  for (int t = 0; t < COT; ++t) acc[t] = (v8f){};

  #pragma unroll
  for (int k0 = 0; k0 < K; k0 += 4) {
    const int ka = k0 + 2 * hi;
    v2f b;
    b.x = load_in<KS>(In, ka,     x, h, H, W, HW);
    b.y = load_in<KS>(In, ka + 1, x, h, H, W, HW);
    #pragma unroll
    for (int t = 0; t < COT; ++t) {
      v2f a;
      a.x = wlds[(t * 16 + row) * K + ka];
      a.y = wlds[(t * 16 + row) * K + ka + 1];
      acc[t] = __builtin_amdgcn_wmma_f32_16x16x4_f32(false, a, false, b,
                                                     (short)0, acc[t], false, false);
    }
  }

  if (x < W) {
    #pragma unroll
    for (int t = 0; t < COT; ++t) {
      #pragma unroll
      for (int r = 0; r < 8; ++r) {
        const int co = t * 16 + r + hi * 8;
        if (co < C) Out[co * HW + h * W + x] = acc[t][r];
      }
    }
  }
}

// ---------------------------------------------------------------------------
// Per-channel mean / rstd over H*W (train-mode BN, N=1, biased variance)
// ---------------------------------------------------------------------------
__global__ void bn_stats_kernel(const float* __restrict__ x, float* __restrict__ stats,
                                int HW) {
  __shared__ float ssum[256];
  __shared__ float ssq[256];
  const int c = blockIdx.x;
  float s = 0.0f, q = 0.0f;
  for (int i = threadIdx.x; i < HW; i += blockDim.x) {
    float v = x[c * HW + i];
    s += v; q += v * v;
  }
  ssum[threadIdx.x] = s; ssq[threadIdx.x] = q;
  __syncthreads();
  for (int off = 128; off > 0; off >>= 1) {
    if ((int)threadIdx.x < off) {
      ssum[threadIdx.x] += ssum[threadIdx.x + off];
      ssq[threadIdx.x]  += ssq[threadIdx.x + off];
    }
    __syncthreads();
  }
  if (threadIdx.x == 0) {
    float m = ssum[0] / (float)HW;
    float v = ssq[0] / (float)HW - m * m;
    stats[2 * c]     = m;
    stats[2 * c + 1] = rsqrtf(v + 1e-5f);
  }
}

// BN normalize + scale/bias (+ optional residual) + activation, in place OK
__global__ void bn_act_kernel(const float* __restrict__ x, const float* __restrict__ stats,
                              const float* __restrict__ g, const float* __restrict__ b,
                              const float* __restrict__ residual, float* __restrict__ out,
                              int HW, int n, int act) {
  int i = blockIdx.x * blockDim.x + threadIdx.x;
  if (i >= n) return;
  int c = i / HW;
  float m = stats[2 * c], r = stats[2 * c + 1];
  float y = g[c] * (x[i] - m) * r + b[c];
  if (residual) y += residual[i];
  if (act == ACT_LEAKY) y = (y >= 0.0f) ? y : 0.2f * y;
  else if (act == ACT_RELU) y = fmaxf(y, 0.0f);
  out[i] = y;
}

// ---------------------------------------------------------------------------
// Correlation volume at 1/16: cv[d,h,w] = mean_c lf[c,h,w]*rf[c,h,w-d], 0 if w<d
// ---------------------------------------------------------------------------
__global__ void corr_kernel(const float* __restrict__ lf, const float* __restrict__ rf,
                            float* __restrict__ cv, int C, int D, int H, int W) {
  int i = blockIdx.x * blockDim.x + threadIdx.x;
  int HW = H * W;
  if (i >= D * HW) return;
  int d = i / HW;
  int p = i - d * HW;
  int w = p % W;
  float s = 0.0f;
  if (w >= d) {
    for (int c = 0; c < C; ++c) s += lf[c * HW + p] * rf[c * HW + p - d];
    s /= (float)C;
  }
  cv[i] = s;
}

// ---------------------------------------------------------------------------
// Bilinear x2 upsample (half-pixel centers), times 2.0
// ---------------------------------------------------------------------------
__global__ void up2_kernel(const float* __restrict__ in, float* __restrict__ out,
                           int H, int W) {
  int Wo = 2 * W;
  int n = 2 * H * Wo;
  int i = blockIdx.x * blockDim.x + threadIdx.x;
  if (i >= n) return;
  int oy = i / Wo;
  int ox = i - oy * Wo;
  float sy = fminf(fmaxf(oy * 0.5f - 0.25f, 0.0f), (float)(H - 1));
  float sx = fminf(fmaxf(ox * 0.5f - 0.25f, 0.0f), (float)(W - 1));
  int y0 = (int)sy, x0 = (int)sx;
  int y1 = min(y0 + 1, H - 1), x1 = min(x0 + 1, W - 1);
  float fy = sy - y0, fx = sx - x0;
  float v = (1.0f - fy) * ((1.0f - fx) * in[y0 * W + x0] + fx * in[y0 * W + x1])
          +          fy * ((1.0f - fx) * in[y1 * W + x0] + fx * in[y1 * W + x1]);
  out[i] = 2.0f * v;
}

// ---------------------------------------------------------------------------
// Sampled cost volume: cand = clip(round(pre)+off, 0, md); 1-D horizontal
// bilinear warp of rf with border clamp; cost = mean_c lf * warp(rf)
// ---------------------------------------------------------------------------
__global__ void sample_cost_kernel(const float* __restrict__ lf, const float* __restrict__ rf,
                                   const float* __restrict__ pre,
                                   float* __restrict__ cost, float* __restrict__ cands,
                                   int C, int S, int mdisp, int H, int W) {
  const int HW = H * W;
  const int n = S * HW;
  int i = blockIdx.x * blockDim.x + threadIdx.x;
  if (i >= n) return;
  int s = i / HW;
  int p = i - s * HW;
  int w = p % W;
  int h = p / W;
  float cand = rintf(pre[p]) + (float)(s - S / 2);
  cand = fminf(fmaxf(cand, 0.0f), (float)mdisp);
  float xs = fminf(fmaxf((float)w - cand, 0.0f), (float)(W - 1));
  float x0f = floorf(xs);
  float wg = xs - x0f;
  int x0 = (int)x0f;
  int x1 = min(x0 + 1, W - 1);
  const float* lrow = lf + h * W;
  const float* rrow = rf + h * W;
  float acc = 0.0f;
  for (int c = 0; c < C; ++c) {
    float r0 = rrow[c * HW + x0];
    float r1 = rrow[c * HW + x1];
    acc += lrow[c * HW + w] * (r0 * (1.0f - wg) + r1 * wg);
  }
  cost[i] = acc / (float)C;
  cands[i] = cand;
}

// ---------------------------------------------------------------------------
// Softmax regression
// ---------------------------------------------------------------------------
__global__ void regress_init_kernel(const float* __restrict__ cv, float* __restrict__ out,
                                    int D, int HW) {
  int i = blockIdx.x * blockDim.x + threadIdx.x;
  if (i >= HW) return;
  float mx = -3.4e38f;
  for (int d = 0; d < D; ++d) mx = fmaxf(mx, cv[d * HW + i]);
  float sum = 0.0f, acc = 0.0f;
  for (int d = 0; d < D; ++d) {
    float e = __expf(cv[d * HW + i] - mx);
    sum += e;
    acc += e * (float)d;
  }
  out[i] = acc / sum;
}

__global__ void regress_sampled_kernel(const float* __restrict__ cv,
                                       const float* __restrict__ cands,
                                       float* __restrict__ out, int S, int HW) {
  int i = blockIdx.x * blockDim.x + threadIdx.x;
  if (i >= HW) return;
  float mx = -3.4e38f;
  for (int s = 0; s < S; ++s) mx = fmaxf(mx, cv[s * HW + i]);
  float sum = 0.0f, acc = 0.0f;
  for (int s = 0; s < S; ++s) {
    float e = __expf(cv[s * HW + i] - mx);
    sum += e;
    acc += e * cands[s * HW + i];
  }
  out[i] = acc / sum;
}

// ---------------------------------------------------------------------------
// Host-side orchestration
// ---------------------------------------------------------------------------
static void conv_bn_act(const float* in, const float* w, const float* g, const float* bsh,
                        const float* residual, float* out, float* stats,
                        int Cc, int H, int W, int ks, int act, hipStream_t stream) {
  const int tilesW = (W + 15) / 16;
  const int waves  = tilesW * H;
  const unsigned blocks = (unsigned)((waves + 3) / 4);
  if (ks == 1) {
    if (Cc == 12) conv_wmma_kernel<1, 12><<<blocks, 128, 0, stream>>>(in, w, out, H, W);
    else          conv_wmma_kernel<1, 24><<<blocks, 128, 0, stream>>>(in, w, out, H, W);
  } else {
    if (Cc == 12) conv_wmma_kernel<3, 12><<<blocks, 128, 0, stream>>>(in, w, out, H, W);
    else          conv_wmma_kernel<3, 24><<<blocks, 128, 0, stream>>>(in, w, out, H, W);
  }
  if (g) {
    const int HW = H * W;
    bn_stats_kernel<<<Cc, 256, 0, stream>>>(out, stats, HW);
    const int n = Cc * HW;
    bn_act_kernel<<<(n + 255) / 256, 256, 0, stream>>>(out, stats, g, bsh, residual,
                                                       out, HW, n, act);
  }
}

// P: 22 tensors of one agg in setup_inputs() insertion order.
// Input cost volume in A; aggregated result left in B.
static void cost_agg_run(const float* const* P, float* A, float* B, float* C,
                         float* stats, int Cc, int H, int W, hipStream_t stream) {
  // conv0: 1x1 + BN + leaky          A -> B        (x' = B, idn = B)
  conv_bn_act(A, P[0], P[1], P[2], nullptr, B, stats, Cc, H, W, 1, ACT_LEAKY, stream);
  // blk0
  conv_bn_act(B, P[3], P[4], P[5], nullptr, C, stats, Cc, H, W, 1, ACT_RELU, stream);
  conv_bn_act(C, P[6], P[7], P[8], nullptr, A, stats, Cc, H, W, 3, ACT_RELU, stream);
  conv_bn_act(A, P[9], P[10], P[11], /*res*/B, C, stats, Cc, H, W, 1, ACT_RELU, stream); // x'' = C
  // blk1
  conv_bn_act(C, P[12], P[13], P[14], nullptr, A, stats, Cc, H, W, 1, ACT_RELU, stream);
  conv_bn_act(A, P[15], P[16], P[17], nullptr, B, stats, Cc, H, W, 3, ACT_RELU, stream);
  conv_bn_act(B, P[18], P[19], P[20], /*res*/C, A, stats, Cc, H, W, 1, ACT_RELU, stream); // x''' = A
  // last conv (no BN)                A -> B
  conv_bn_act(A, P[21], nullptr, nullptr, nullptr, B, stats, Cc, H, W, 1, ACT_NONE, stream);
}

extern "C" void kernel_launch(void* const* d_in, const int* in_sizes, int n_in,
                              void* d_out, int out_size, void* d_ws, size_t ws_size,
                              hipStream_t stream) {
  (void)in_sizes; (void)n_in; (void)out_size; (void)ws_size;
  const int H16 = 24,  W16 = 80;
  const int H8  = 48,  W8  = 160;
  const int H4  = 96,  W4  = 320;
  const int H2  = 192, W2  = 640;
  const int H1  = 384, W1  = 1280;

  // Feature inputs (setup_inputs insertion order)
  const float* lf16 = (const float*)d_in[0];
  const float* lf8  = (const float*)d_in[1];
  const float* lf4  = (const float*)d_in[2];
  const float* lf2  = (const float*)d_in[3];
  const float* lf1  = (const float*)d_in[4];
  const float* rf16 = (const float*)d_in[5];
  const float* rf8  = (const float*)d_in[6];
  const float* rf4  = (const float*)d_in[7];
  const float* rf2  = (const float*)d_in[8];
  const float* rf1  = (const float*)d_in[9];

  // params: aggs in order agg16, agg8, agg4, agg2, agg1; each agg = 22 tensors:
  // conv0_w,conv0_g,conv0_b, blk0{w1,g1,b1,w2,g2,b2,w3,g3,b3}, blk1{...}, last_w
  const float* P16[22]; const float* P8[22]; const float* P4[22];
  const float* P2[22];  const float* P1[22];
  {
    const float** Ps[5] = {P16, P8, P4, P2, P1};
    for (int j = 0; j < 5; ++j)
      for (int t = 0; t < 22; ++t)
        Ps[j][t] = (const float*)d_in[10 + j * 22 + t];
  }

  // Workspace layout (floats)
  const long bufElems = 12L * H1 * W1;        // 5,898,240 (largest C*HW over all stages)
  float* ws = (float*)d_ws;
  float* A     = ws;
  float* B     = A + bufElems;
  float* C     = B + bufElems;
  float* CANDS = C + bufElems;
  float* PRE   = CANDS + bufElems;
  float* STATS = PRE + (long)H1 * W1;

  // Output slices (disp16, disp8, disp4, disp2, disp1 concatenated)
  float* out16 = (float*)d_out;
  float* out8  = out16 + (long)H16 * W16;
  float* out4  = out8  + (long)H8  * W8;
  float* out2  = out4  + (long)H4  * W4;
  float* out1  = out2  + (long)H2  * W2;

  // ---- scale 1/16: correlation volume + agg + init regression ----
  {
    const int D = 12, HW = H16 * W16;
    corr_kernel<<<(D * HW + 255) / 256, 256, 0, stream>>>(lf16, rf16, A, 32, D, H16, W16);
    cost_agg_run(P16, A, B, C, STATS, 12, H16, W16, stream);
    regress_init_kernel<<<(HW + 255) / 256, 256, 0, stream>>>(B, out16, D, HW);
  }

  struct Stage {
    const float *lf, *rf;
    const float* const* P;
    int Cf, S, md, Hp, Wp, H, W, Cagg;
    float* out;
    const float* prevDisp;
  };
  Stage stages[4] = {
    { lf8, rf8, P8, 24, 24, 24,  H16, W16, H8, W8, 24, out8, out16 },
    { lf4, rf4, P4, 24, 24, 48,  H8,  W8,  H4, W4, 24, out4, out8  },
    { lf2, rf2, P2, 16, 24, 96,  H4,  W4,  H2, W2, 24, out2, out4  },
    { lf1, rf1, P1, 16, 12, 192, H2,  W2,  H1, W1, 12, out1, out2  },
  };

  for (int st = 0; st < 4; ++st) {
    const Stage& sg = stages[st];
    const int HW = sg.H * sg.W;
    // pre-disparity: bilinear x2 upsample of previous disp, times 2
    up2_kernel<<<(HW + 255) / 256, 256, 0, stream>>>(sg.prevDisp, PRE, sg.Hp, sg.Wp);
    // sampled cost volume into A (+ candidate disparities into CANDS)
    const int n = sg.S * HW;
    sample_cost_kernel<<<(n + 255) / 256, 256, 0, stream>>>(
        sg.lf, sg.rf, PRE, A, CANDS, sg.Cf, sg.S, sg.md, sg.H, sg.W);
    // aggregation (WMMA convs) -> B
    cost_agg_run(sg.P, A, B, C, STATS, sg.Cagg, sg.H, sg.W, stream);
    // softmax regression over candidates
    regress_sampled_kernel<<<(HW + 255) / 256, 256, 0, stream>>>(
        B, CANDS, sg.out, sg.S, HW);
  }
}